// MLP_CRF_NER_39075612459436
// MI455X (gfx1250) — compile-verified
//
#include <hip/hip_runtime.h>
#include <hip/hip_bf16.h>
#include <stdint.h>

// ---------------------------------------------------------------------------
// MLP + CRF NER negative log-likelihood for MI455X (gfx1250, wave32).
//
// Sizes (fixed by reference): H=512, L=21, B=128, S=512.
//   1) init_pad:   pad W (21x512) -> Wpad (32x512, zero rows), b -> bpad(32)
//   2) emis_gemm:  emis[b,s,l] = fv[b,s,:] . W[l,:] + b[l] with
//                  V_WMMA_F32_16X16X4_F32. A tiles (128x32 f32) are DMA'd
//                  memory->LDS by the Tensor Data Mover (tensor_load_to_lds),
//                  double-buffered and synced with s_wait_tensorcnt. TDM's
//                  LDS pad feature (4 DW pad every 32 DW) gives the
//                  conflict-free 36-float row stride for free.
//                  Memory-bound: 134 MB in -> ~6us roofline at 23.3 TB/s.
//   3) crf_kernel: one wave32 per batch element; 511 serial logsumexp steps,
//                  alpha broadcast via __shfl, max/sum trees on the critical
//                  path. Score term lane-parallel + wave reduction.
//   4) final_sum:  single-thread fixed-order sum (deterministic).
// ---------------------------------------------------------------------------

#define H_DIM 512
#define L_NUM 21
#define L_PAD 32
#define B_NUM 128
#define S_LEN 512
#define BS_ROWS (B_NUM * S_LEN)   // 65536
#define BK 32
#define NK_ITERS (H_DIM / BK)     // 16
#define LDS_STRIDE 36             // 32 data DW + 4 pad DW (TDM pad_amount)

typedef __attribute__((ext_vector_type(2))) float v2f;
typedef __attribute__((ext_vector_type(8))) float v8f;
typedef __attribute__((ext_vector_type(4))) unsigned int v4u;
typedef __attribute__((ext_vector_type(8))) int v8i;
typedef __attribute__((ext_vector_type(4))) int v4i;

// ---------------------------------------------------------------------------
// Kernel 1: pad W and bias into workspace
// ---------------------------------------------------------------------------
__global__ void init_pad_kernel(const float* __restrict__ W,
                                const float* __restrict__ bias,
                                float* __restrict__ Wpad,
                                float* __restrict__ bpad) {
    int idx = blockIdx.x * blockDim.x + threadIdx.x;   // 0 .. 16383
    if (idx < L_PAD * H_DIM) {
        int n = idx / H_DIM;
        int k = idx % H_DIM;
        Wpad[idx] = (n < L_NUM) ? W[n * H_DIM + k] : 0.0f;
    }
    if (idx < L_PAD) {
        bpad[idx] = (idx < L_NUM) ? bias[idx] : 0.0f;
    }
}

// ---------------------------------------------------------------------------
// TDM: issue one tensor_load_to_lds moving a 128x32-f32 tile (row stride
// H_DIM) from global memory into LDS at lds_byte_off, padding the LDS
// destination by 4 DWORDs after every 32 DWORDs (-> 36-float row stride).
// D# bit layout per CDNA5 ISA 8.3/8.4.
// ---------------------------------------------------------------------------
__device__ __forceinline__ void tdm_load_tile(uint32_t lds_byte_off,
                                              const float* gptr) {
    const uint64_t ga = (uint64_t)(uintptr_t)gptr;
    v4u g0;
    g0.x = 1u;                                          // count=1 (valid D#)
    g0.y = lds_byte_off;                                // lds_addr
    g0.z = (uint32_t)(ga & 0xFFFFFFFFu);                // global_addr[31:0]
    g0.w = (uint32_t)((ga >> 32) & 0x01FFFFFFu)         // global_addr[56:32]
         | (2u << 30);                                  // type=2 ("image")
    v8i g1;
    g1[0] = (int)((2u << 16)        // data_size = 4 bytes
                | (1u << 20)        // pad_enable
                | (4u << 22)        // pad_interval: every 32 DWORDs
                | (3u << 25));      // pad_amount: 4 DWORDs
    g1[1] = (int)(512u << 16);      // tensor_dim0[15:0] = 512
    g1[2] = 0;                      // tensor_dim0 hi / tensor_dim1 lo (65536)
    g1[3] = (int)((32u << 16) | 1u);// tensor_dim1 hi =1 ; tile_dim0 = 32
    g1[4] = 128;                    // tile_dim1 = 128 ; tile_dim2 = 0
    g1[5] = (int)H_DIM;             // tensor_dim0_stride = 512 elements
    g1[6] = 0;
    g1[7] = 0;
    v4i z4 = {0, 0, 0, 0};
#if __has_include(<hip/amd_detail/amd_gfx1250_TDM.h>)
    // amdgpu-toolchain (clang-23 / therock): 6-arg builtin
    v8i z8 = {0, 0, 0, 0, 0, 0, 0, 0};
    __builtin_amdgcn_tensor_load_to_lds(g0, g1, z4, z4, z8, 0);
#else
    // ROCm 7.2 (clang-22): 5-arg builtin
    __builtin_amdgcn_tensor_load_to_lds(g0, g1, z4, z4, 0);
#endif
}

// ---------------------------------------------------------------------------
// Kernel 2: emissions GEMM via V_WMMA_F32_16X16X4_F32 + TDM staging.
//   Block: 256 threads = 8 waves; block M-tile = 128 rows; wave M-tile = 16.
//   N: 21 padded to 32 -> two 16-wide accumulators per wave.
//   K loop: chunks of 32, TDM double-buffered in LDS.
// WMMA f32 16x16x4 fragment layout (ISA 7.12.2):
//   A (16x4): lane m / m+16: VGPR j holds A[m][2*half + j]
//   B (4x16): lane n / n+16: VGPR j holds B[2*half + j][n]
//   C (16x16): VGPR v: lanes 0-15 -> M=v, lanes 16-31 -> M=v+8, N=lane%16
// ---------------------------------------------------------------------------
__global__ __launch_bounds__(256) void emis_gemm_kernel(
    const float* __restrict__ fv,      // [BS_ROWS, H]
    const float* __restrict__ Wpad,    // [32, H]
    const float* __restrict__ bpad,    // [32]
    float* __restrict__ emis)          // [BS_ROWS, 32]
{
    __shared__ __align__(16) float As[2][128 * LDS_STRIDE];

    const int tid  = threadIdx.x;
    const int wave = tid >> 5;
    const int lane = tid & 31;
    const int m    = lane & 15;
    const int hlf  = lane >> 4;
    const int row_base = blockIdx.x * 128;

    const uint32_t lds_base = (uint32_t)(uintptr_t)(&As[0][0]);
    const uint32_t lds_buf_stride = 128u * LDS_STRIDE * 4u;
    const float* gtile = fv + (size_t)row_base * H_DIM;

    v8f acc0 = {0.f, 0.f, 0.f, 0.f, 0.f, 0.f, 0.f, 0.f};
    v8f acc1 = {0.f, 0.f, 0.f, 0.f, 0.f, 0.f, 0.f, 0.f};

    // Preload K-chunk 0 into buffer 0 (one TDM op per block, issued by wave 0)
    if (wave == 0) {
        tdm_load_tile(lds_base, gtile);
    }

    for (int i = 0; i < NK_ITERS; ++i) {
        const int buf = i & 1;
        if (wave == 0) {
            if (i + 1 < NK_ITERS) {
                // Prefetch next K-chunk into the other buffer, then wait for
                // chunk i (TDM ops from one wave complete in order).
                tdm_load_tile(lds_base + (uint32_t)((i + 1) & 1) * lds_buf_stride,
                              gtile + (size_t)(i + 1) * BK);
                __builtin_amdgcn_s_wait_tensorcnt(1);
            } else {
                __builtin_amdgcn_s_wait_tensorcnt(0);
            }
        }
        __syncthreads();   // chunk i visible in LDS to all waves

        const int k0 = i * BK;
        #pragma unroll
        for (int kk = 0; kk < BK; kk += 4) {
            // A fragment from LDS (8B-aligned ds_load_b64, conflict-free)
            v2f a = *(const v2f*)(&As[buf][(wave * 16 + m) * LDS_STRIDE + kk + 2 * hlf]);
            // B fragments: B[k][n] = Wpad[n][k]; W is tiny (64KB) & cache-hot
            v2f b0 = *(const v2f*)(Wpad + (size_t)m        * H_DIM + k0 + kk + 2 * hlf);
            v2f b1 = *(const v2f*)(Wpad + (size_t)(m + 16) * H_DIM + k0 + kk + 2 * hlf);
            // 8 args: (neg_a, A, neg_b, B, c_mod, C, reuse_a, reuse_b)
            acc0 = __builtin_amdgcn_wmma_f32_16x16x4_f32(
                false, a, false, b0, (short)0, acc0, false, false);
            acc1 = __builtin_amdgcn_wmma_f32_16x16x4_f32(
                false, a, false, b1, (short)0, acc1, false, false);
        }
        __syncthreads();   // all waves done with buf before TDM overwrites it
    }

    // Epilogue: add bias, store padded emissions (64B-coalesced segments)
    const float bia0 = bpad[m];
    const float bia1 = bpad[m + 16];
    #pragma unroll
    for (int v = 0; v < 8; ++v) {
        int row = row_base + wave * 16 + v + 8 * hlf;
        emis[(size_t)row * L_PAD + m]      = acc0[v] + bia0;
        emis[(size_t)row * L_PAD + m + 16] = acc1[v] + bia1;
    }
}

// ---------------------------------------------------------------------------
// Tree reductions over 21 register values (shorten the serial critical path)
// ---------------------------------------------------------------------------
__device__ __forceinline__ float tree_max21(const float* v) {
    float r[11];
    #pragma unroll
    for (int i = 0; i < 10; ++i) r[i] = fmaxf(v[i], v[i + 10]);
    r[10] = v[20];
    float s[6];
    #pragma unroll
    for (int i = 0; i < 5; ++i) s[i] = fmaxf(r[i], r[i + 5]);
    s[5] = r[10];
    float t0 = fmaxf(s[0], s[3]);
    float t1 = fmaxf(s[1], s[4]);
    float t2 = fmaxf(s[2], s[5]);
    return fmaxf(fmaxf(t0, t1), t2);
}

__device__ __forceinline__ float tree_sum21(const float* v) {
    float r[11];
    #pragma unroll
    for (int i = 0; i < 10; ++i) r[i] = v[i] + v[i + 10];
    r[10] = v[20];
    float s[6];
    #pragma unroll
    for (int i = 0; i < 5; ++i) s[i] = r[i] + r[i + 5];
    s[5] = r[10];
    return ((s[0] + s[3]) + (s[1] + s[4])) + (s[2] + s[5]);
}

// ---------------------------------------------------------------------------
// Kernel 3: CRF forward + gold score. One wave32 per batch element.
//   mask is a JAX bool array -> 1 byte per element on device.
// ---------------------------------------------------------------------------
__global__ __launch_bounds__(32) void crf_kernel(
    const float* __restrict__ emis,          // [BS_ROWS, 32] padded
    const int* __restrict__ labels,          // [B, S]
    const unsigned char* __restrict__ mask,  // [B, S] bool
    const float* __restrict__ start_trans,   // [L]
    const float* __restrict__ end_trans,     // [L]
    const float* __restrict__ trans,         // [L, L]
    float* __restrict__ partial)             // [B]
{
    const int b    = blockIdx.x;
    const int lane = threadIdx.x;
    const bool act = (lane < L_NUM);
    const size_t base = (size_t)b * S_LEN;

    // Lane j keeps column j of trans in registers: tc[i] = trans[i][j]
    float tc[L_NUM];
    #pragma unroll
    for (int i = 0; i < L_NUM; ++i)
        tc[i] = act ? trans[i * L_NUM + lane] : 0.0f;

    // alpha0 = start_trans + emis[t=0]
    float alpha = act ? (start_trans[lane] + emis[base * L_PAD + lane]) : -1e30f;

    // ---- forward recursion (serial over t; latency-bound) ----
    for (int t = 1; t < S_LEN; ++t) {
        float e = act ? emis[(base + t) * L_PAD + lane] : 0.0f;
        float vals[L_NUM];
        #pragma unroll
        for (int i = 0; i < L_NUM; ++i) {
            float ai = __shfl(alpha, i, 32);     // 21 independent broadcasts
            vals[i] = ai + tc[i];
        }
        float mx = tree_max21(vals);
        float ex[L_NUM];
        #pragma unroll
        for (int i = 0; i < L_NUM; ++i) ex[i] = __expf(vals[i] - mx);
        float nxt = mx + __logf(tree_sum21(ex)) + e;
        alpha = (mask[base + t] != 0) ? nxt : alpha;
    }

    // ---- logZ = logsumexp(alpha + end_trans) across lanes 0..20 ----
    float x  = act ? (alpha + end_trans[lane]) : -1e30f;
    float mx = x;
    #pragma unroll
    for (int off = 16; off > 0; off >>= 1)
        mx = fmaxf(mx, __shfl_xor(mx, off, 32));
    float ex = act ? __expf(x - mx) : 0.0f;
    #pragma unroll
    for (int off = 16; off > 0; off >>= 1)
        ex += __shfl_xor(ex, off, 32);
    float logZ = mx + __logf(ex);

    // ---- gold-path score: lane-parallel over timesteps ----
    float part = 0.0f, mcnt = 0.0f;
    for (int t = lane; t < S_LEN; t += 32) {
        float mt = (mask[base + t] != 0) ? 1.0f : 0.0f;
        mcnt += mt;
        if (t >= 1) {
            int lp = labels[base + t - 1];
            int lc = labels[base + t];
            part += mt * (trans[lp * L_NUM + lc] + emis[(base + t) * L_PAD + lc]);
        }
    }
    #pragma unroll
    for (int off = 16; off > 0; off >>= 1) {
        part += __shfl_xor(part, off, 32);
        mcnt += __shfl_xor(mcnt, off, 32);
    }

    if (lane == 0) {
        int l0   = labels[base];
        int last = (int)mcnt - 1;                // sum(mask) - 1
        int ll   = labels[base + last];
        float score = start_trans[l0] + emis[base * L_PAD + l0]
                    + part + end_trans[ll];
        partial[b] = logZ - score;
    }
}

// ---------------------------------------------------------------------------
// Kernel 4: deterministic fixed-order final sum
// ---------------------------------------------------------------------------
__global__ void final_sum_kernel(const float* __restrict__ partial,
                                 float* __restrict__ out) {
    if (threadIdx.x == 0 && blockIdx.x == 0) {
        float s = 0.0f;
        for (int i = 0; i < B_NUM; ++i) s += partial[i];
        out[0] = s;
    }
}

// ---------------------------------------------------------------------------
// Host-side launch
// Inputs (setup_inputs order): fused_vectors f32 [B,S,H], labels i32 [B,S],
//   mask bool [B,S], W f32 [L,H], b f32 [L], start_trans f32 [L],
//   end_trans f32 [L], trans f32 [L,L]. Output: scalar f32.
// Workspace layout (needs ~8.6 MB):
//   [0, 64KB)          Wpad  32x512 f32
//   [64KB, 64KB+128)   bpad  32 f32
//   [64KB+128, ...)    partial 128 f32
//   [128KB, 128KB+8MB) emis 65536x32 f32
// ---------------------------------------------------------------------------
extern "C" void kernel_launch(void* const* d_in, const int* in_sizes, int n_in,
                              void* d_out, int out_size, void* d_ws, size_t ws_size,
                              hipStream_t stream) {
    const float*         fv     = (const float*)d_in[0];
    const int*           labels = (const int*)d_in[1];
    const unsigned char* mask   = (const unsigned char*)d_in[2];
    const float*         W      = (const float*)d_in[3];
    const float*         bias   = (const float*)d_in[4];
    const float*         st     = (const float*)d_in[5];
    const float*         et     = (const float*)d_in[6];
    const float*         tr     = (const float*)d_in[7];
    float* out = (float*)d_out;

    char*  ws      = (char*)d_ws;
    float* Wpad    = (float*)(ws);
    float* bpad    = (float*)(ws + 65536);
    float* partial = (float*)(ws + 65536 + 128);
    float* emis    = (float*)(ws + 131072);

    init_pad_kernel<<<64, 256, 0, stream>>>(W, bias, Wpad, bpad);
    emis_gemm_kernel<<<BS_ROWS / 128, 256, 0, stream>>>(fv, Wpad, bpad, emis);
    crf_kernel<<<B_NUM, 32, 0, stream>>>(emis, labels, mask, st, et, tr, partial);
    final_sum_kernel<<<1, 32, 0, stream>>>(partial, out);
}